// PhaseEncoding_57672820850799
// MI455X (gfx1250) — compile-verified
//
#include <hip/hip_runtime.h>

// PhaseEncoding: x (8,4096,512) f32 -> one-hot over 8 phase bins (…,8) f32.
// Store-bandwidth-bound: ~604 MB traffic @ 23.3 TB/s => ~26 us floor.
// Strategy: 1x b128 NT load + 8x b128 NT stores per thread, wave32, bins
// staged to LDS via the CDNA5 async global->LDS path.

#define PE_TWO_PI     6.28318530717958647692f
#define PE_INV_TWO_PI 0.15915494309189533577f

typedef float pe_float4 __attribute__((ext_vector_type(4)));
typedef __attribute__((address_space(1))) int* pe_gptr;
typedef __attribute__((address_space(3))) int* pe_lptr;

#if __has_builtin(__builtin_amdgcn_global_load_async_to_lds_b32) && \
    __has_builtin(__builtin_amdgcn_s_wait_asynccnt)
#define PE_HAS_ASYNC_LDS 1
#else
#define PE_HAS_ASYNC_LDS 0
#endif

__global__ __launch_bounds__(256) void PhaseEncoding_57672820850799_kernel(
    const float* __restrict__ x,
    const float* __restrict__ bins,   // 9 floats: linspace(0, 2pi, 9)
    float* __restrict__ out,
    int nvec)                         // number of float4 chunks of x
{
    __shared__ float sbins[16];       // 9 used, padded for alignment
    const int t = threadIdx.x;

#if PE_HAS_ASYNC_LDS
    // CDNA5 async global->LDS broadcast of the tiny bin table (9 dwords).
    if (t < 9) {
        __builtin_amdgcn_global_load_async_to_lds_b32(
            (pe_gptr)(bins + t),
            (pe_lptr)(&sbins[t]),
            /*offset=*/0, /*cpol=*/0);
    }
    __builtin_amdgcn_s_wait_asynccnt(0);
#else
    if (t < 9) sbins[t] = bins[t];
#endif
    __syncthreads();

    float b[9];
#pragma unroll
    for (int i = 0; i < 9; ++i) b[i] = sbins[i];

    const int gid = blockIdx.x * blockDim.x + t;
    if (gid >= nvec) return;

    const pe_float4* __restrict__ xv = (const pe_float4*)x;
    pe_float4 v = __builtin_nontemporal_load(xv + gid);

    pe_float4* __restrict__ ov = (pe_float4*)out;

#pragma unroll
    for (int e = 0; e < 4; ++e) {
        const float xe = v[e];
        // x >= 0 here, so trunc == floor: exact x mod 2pi for this domain.
        const float xn = __builtin_fmaf(-PE_TWO_PI,
                                        truncf(xe * PE_INV_TWO_PI), xe);
        pe_float4 lo, hi;
#pragma unroll
        for (int p = 0; p < 4; ++p)
            lo[p] = (xn >= b[p]     && xn < b[p + 1]) ? 1.0f : 0.0f;
#pragma unroll
        for (int p = 0; p < 4; ++p)
            hi[p] = (xn >= b[p + 4] && xn < b[p + 5]) ? 1.0f : 0.0f;

        // Each input element owns 8 contiguous output floats = 2 float4s.
        const size_t base = (size_t)gid * 8u + (size_t)e * 2u;
        __builtin_nontemporal_store(lo, ov + base);
        __builtin_nontemporal_store(hi, ov + base + 1);
    }
}

extern "C" void kernel_launch(void* const* d_in, const int* in_sizes, int n_in,
                              void* d_out, int out_size, void* d_ws, size_t ws_size,
                              hipStream_t stream) {
    const float* x    = (const float*)d_in[0];
    const float* bins = (const float*)d_in[1];
    float* out        = (float*)d_out;

    const int n    = in_sizes[0];     // 8*4096*512 = 16,777,216 (divisible by 4)
    const int nvec = n / 4;           // 4,194,304 float4 chunks

    const int block = 256;            // 8 wave32 waves per block
    const int grid  = (nvec + block - 1) / block;   // 16384 blocks

    PhaseEncoding_57672820850799_kernel<<<grid, block, 0, stream>>>(x, bins, out, nvec);
}